// Transition_33809982554700
// MI455X (gfx1250) — compile-verified
//
#include <hip/hip_runtime.h>

typedef float v2f __attribute__((ext_vector_type(2)));
typedef float v8f __attribute__((ext_vector_type(8)));

#define N_DIM   2000
#define C_DIM   32
#define J_DIM   4
#define BS_DIM  4
#define NCH     128          // J*C
#define PER_CH  8000.0f      // BS*N elements per channel for BN stats
#define BN_EPS  1e-5f

// fp32 WMMA: D(16x16 f32) = A(16x4 f32) * B(4x16 f32) + C
__device__ __forceinline__ v8f wmma4(v2f a, v2f b, v8f c) {
    return __builtin_amdgcn_wmma_f32_16x16x4_f32(
        /*neg_a=*/false, a, /*neg_b=*/false, b,
        /*c_mod=*/(short)0, c, /*reuse_a=*/false, /*reuse_b=*/false);
}

// ---- workspace layout (floats) ----
// [0, 1024000)            y  : 4 x 128 x 2000
// [1024000, 1024128)      sum per channel
// [1024128, 1024256)      sumsq per channel
// [1024256, 1024384)      scale per channel
// [1024384, 1024512)      shift per channel
#define WS_Y      0
#define WS_SUM    1024000
#define WS_SQ     1024128
#define WS_SCALE  1024256
#define WS_SHIFT  1024384

__global__ void k_init(float* __restrict__ gsum, float* __restrict__ gsq) {
    int t = threadIdx.x;
    if (t < NCH) { gsum[t] = 0.0f; gsq[t] = 0.0f; }
}

// grid = (16, 16): blockIdx.x covers 128 M-tiles (125 used), blockIdx.y = b*4+j
// block = 256 threads (8 waves); each wave owns one 16-row M-tile, both C tiles.
__global__ void k_gmul(const float* __restrict__ W, const float* __restrict__ x,
                       float* __restrict__ y, float* __restrict__ gsum,
                       float* __restrict__ gsq) {
    __shared__ float ls[C_DIM];
    __shared__ float lq[C_DIM];
    const int tid  = threadIdx.x;
    const int lane = tid & 31;
    const int wave = tid >> 5;
    const int bj   = blockIdx.y;
    const int b    = bj >> 2;
    const int j    = bj & 3;

    if (tid < C_DIM) { ls[tid] = 0.0f; lq[tid] = 0.0f; }
    __syncthreads();

    const int mtile = blockIdx.x * 8 + wave;   // wave-uniform branch keeps EXEC full
    if (mtile < 125) {
        const int lo = lane & 15;
        const int hi = lane >> 4;
        const int m  = mtile * 16 + lo;
        // A fragment source: W[b][j][m][k + hi*2 + {0,1}]  (single-use -> NT loads)
        const float* pa  = W + (size_t)bj * N_DIM * N_DIM + (size_t)m * N_DIM + hi * 2;
        // B fragment source: x[b][c][k + hi*2 + {0,1}], c = lo (+16 for tile 1)
        const float* pb0 = x + ((size_t)b * C_DIM + lo) * N_DIM + hi * 2;
        const float* pb1 = pb0 + (size_t)16 * N_DIM;

        v8f acc0 = {};
        v8f acc1 = {};
        #pragma unroll 8
        for (int k = 0; k < N_DIM; k += 4) {
            v2f a  = __builtin_nontemporal_load((const v2f*)(pa + k));
            v2f b0 = *(const v2f*)(pb0 + k);
            v2f b1 = *(const v2f*)(pb1 + k);
            acc0 = wmma4(a, b0, acc0);
            acc1 = wmma4(a, b1, acc1);
        }

        // C/D layout: lane holds column c=lo; VGPR r -> row m0+r (m0 has +8 for hi half)
        const int m0  = mtile * 16 + hi * 8;
        const int ch0 = j * C_DIM + lo;        // channel of acc0
        float* yp0 = y + ((size_t)(b * NCH + ch0))      * N_DIM + m0;
        float* yp1 = y + ((size_t)(b * NCH + ch0 + 16)) * N_DIM + m0;
        *(float4*)(yp0)     = make_float4(acc0[0], acc0[1], acc0[2], acc0[3]);
        *(float4*)(yp0 + 4) = make_float4(acc0[4], acc0[5], acc0[6], acc0[7]);
        *(float4*)(yp1)     = make_float4(acc1[0], acc1[1], acc1[2], acc1[3]);
        *(float4*)(yp1 + 4) = make_float4(acc1[4], acc1[5], acc1[6], acc1[7]);

        // BN partial stats: per-lane reduce over 8 rows (channel constant per lane)
        float s0 = 0.f, q0 = 0.f, s1 = 0.f, q1 = 0.f;
        #pragma unroll
        for (int r = 0; r < 8; ++r) {
            s0 += acc0[r]; q0 += acc0[r] * acc0[r];
            s1 += acc1[r]; q1 += acc1[r] * acc1[r];
        }
        // combine lane l with lane l^16 (same channel, other row-half)
        s0 += __shfl_xor(s0, 16); q0 += __shfl_xor(q0, 16);
        s1 += __shfl_xor(s1, 16); q1 += __shfl_xor(q1, 16);
        if (lane < 16) {
            atomicAdd(&ls[lo],      s0); atomicAdd(&lq[lo],      q0);
            atomicAdd(&ls[lo + 16], s1); atomicAdd(&lq[lo + 16], q1);
        }
    }
    __syncthreads();
    if (tid < C_DIM) {
        atomicAdd(&gsum[j * C_DIM + tid], ls[tid]);
        atomicAdd(&gsq [j * C_DIM + tid], lq[tid]);
    }
}

__global__ void k_stats(const float* __restrict__ gsum, const float* __restrict__ gsq,
                        const float* __restrict__ gamma, const float* __restrict__ beta,
                        float* __restrict__ scale, float* __restrict__ shift) {
    int c = threadIdx.x;
    if (c < NCH) {
        float mean = gsum[c] * (1.0f / PER_CH);
        float var  = gsq[c]  * (1.0f / PER_CH) - mean * mean;
        float sc   = gamma[c] * rsqrtf(var + BN_EPS);
        scale[c] = sc;
        shift[c] = beta[c] - mean * sc;
    }
}

// grid = (125, 8): blockIdx.x = n-tile, blockIdx.y = b*2 + o-tile; block = 1 wave
// out[b][o][n] = cb[o] + sum_ch cw[o][ch] * relu(y[b][ch][n]*scale[ch]+shift[ch])
__global__ void k_conv(const float* __restrict__ y, const float* __restrict__ scale,
                       const float* __restrict__ shift, const float* __restrict__ cw,
                       const float* __restrict__ cb, float* __restrict__ out) {
    __shared__ float ssc[NCH];
    __shared__ float ssh[NCH];
    const int lane = threadIdx.x;
    for (int i = lane; i < NCH; i += 32) { ssc[i] = scale[i]; ssh[i] = shift[i]; }
    __syncthreads();

    const int ntile = blockIdx.x;
    const int b     = blockIdx.y >> 1;
    const int ot    = blockIdx.y & 1;
    const int lo    = lane & 15;
    const int hi    = lane >> 4;
    const int o     = ot * 16 + lo;

    const float* pa = cw + (size_t)o * NCH + hi * 2;                  // A: conv_w row
    const float* py = y + (size_t)b * NCH * N_DIM + ntile * 16 + lo;  // B: y column

    v8f acc = {};
    #pragma unroll 4
    for (int k = 0; k < NCH; k += 4) {
        const int ch = k + hi * 2;
        v2f a = *(const v2f*)(pa + k);
        float v0 = __builtin_nontemporal_load(py + (size_t)ch       * N_DIM);
        float v1 = __builtin_nontemporal_load(py + (size_t)(ch + 1) * N_DIM);
        v0 = fmaxf(fmaf(v0, ssc[ch],     ssh[ch]),     0.0f);
        v1 = fmaxf(fmaf(v1, ssc[ch + 1], ssh[ch + 1]), 0.0f);
        v2f bb; bb[0] = v0; bb[1] = v1;
        acc = wmma4(a, bb, acc);
    }

    const int obase = ot * 16 + hi * 8;
    const int n0    = ntile * 16 + lo;
    #pragma unroll
    for (int r = 0; r < 8; ++r) {
        out[((size_t)(b * C_DIM + obase + r)) * N_DIM + n0] = acc[r] + cb[obase + r];
    }
}

extern "C" void kernel_launch(void* const* d_in, const int* in_sizes, int n_in,
                              void* d_out, int out_size, void* d_ws, size_t ws_size,
                              hipStream_t stream) {
    (void)in_sizes; (void)n_in; (void)out_size; (void)ws_size;
    const float* W      = (const float*)d_in[0];  // 4*4*2000*2000
    const float* x      = (const float*)d_in[1];  // 4*32*2000
    const float* gamma  = (const float*)d_in[2];  // 128
    const float* beta   = (const float*)d_in[3];  // 128
    const float* conv_w = (const float*)d_in[4];  // 32*128
    const float* conv_b = (const float*)d_in[5];  // 32
    float* out = (float*)d_out;                   // 4*32*2000

    float* ws     = (float*)d_ws;
    float* y      = ws + WS_Y;
    float* gsum   = ws + WS_SUM;
    float* gsq    = ws + WS_SQ;
    float* sscale = ws + WS_SCALE;
    float* sshift = ws + WS_SHIFT;

    k_init<<<1, 256, 0, stream>>>(gsum, gsq);
    dim3 g1(16, 16);
    k_gmul<<<g1, 256, 0, stream>>>(W, x, y, gsum, gsq);
    k_stats<<<1, 128, 0, stream>>>(gsum, gsq, gamma, beta, sscale, sshift);
    dim3 g3(125, 8);
    k_conv<<<g3, 32, 0, stream>>>(y, sscale, sshift, conv_w, conv_b, out);
}